// TopologicalLoss_66941360275811
// MI455X (gfx1250) — compile-verified
//
#include <hip/hip_runtime.h>

typedef __attribute__((ext_vector_type(2))) float v2f;
typedef __attribute__((ext_vector_type(8))) float v8f;
typedef __attribute__((ext_vector_type(4))) unsigned int v4u;
typedef __attribute__((ext_vector_type(4))) int v4i;
typedef __attribute__((ext_vector_type(8))) int v8i;

#define NN 4096
#define DD 512
#define BIGF 1e30f
// LDS A-strip row stride: 512 data floats + 2 pad dwords per 256 (TDM pad_enable)
#define AStride 516

__device__ __forceinline__ unsigned long long shfl_xor_u64(unsigned long long v, int mask) {
  unsigned lo = (unsigned)v, hi = (unsigned)(v >> 32);
  lo = __shfl_xor(lo, mask, 32);
  hi = __shfl_xor(hi, mask, 32);
  return ((unsigned long long)hi << 32) | lo;
}

// ---------------- row squared norms: one wave per row ----------------
__global__ void sqnorm_kernel(const float* __restrict__ x, float* __restrict__ sq) {
  const int wave = threadIdx.x >> 5;
  const int lane = threadIdx.x & 31;
  const int row  = blockIdx.x * 8 + wave;
  const float* p = x + (size_t)row * DD;
  float s = 0.f;
#pragma unroll
  for (int t = 0; t < DD / 32; ++t) {
    float v = p[lane + t * 32];
    s = fmaf(v, v, s);
  }
#pragma unroll
  for (int off = 16; off > 0; off >>= 1) s += __shfl_xor(s, off, 32);
  if (lane == 0) sq[row] = s;
}

// ---------------- distance matrix: TDM-staged A strip + fp32 WMMA ----------------
// Block = 8 waves; all waves share the 16-row A strip -> stage it once via the
// Tensor Data Mover into LDS (padded stride 516 => conflict-free ds_load_b64).
// Each wave computes a 16x64 tile (4 accumulators reuse one A fragment).
__global__ __launch_bounds__(256) void dist_kernel(const float* __restrict__ x,
                                                   const float* __restrict__ sq,
                                                   float* __restrict__ dist) {
  __shared__ float sA[16 * AStride];   // 33 KB

  const int wave = threadIdx.x >> 5;
  const int lane = threadIdx.x & 31;
  const int hv   = lane >> 4;          // K-half selector
  const int l    = lane & 15;
  const int rowBase = blockIdx.y * 16;
  const int colBase = blockIdx.x * 512 + wave * 64;

  // ---- TDM descriptor: 2D tile 512(x) x 16(y) fp32, pad 2 dwords / 256 dwords ----
  {
    unsigned long long ga = (unsigned long long)(uintptr_t)(x + (size_t)rowBase * DD);
    unsigned ldsOff = (unsigned)(uintptr_t)&sA[0];   // flat LDS addr[31:0] == LDS byte offset
    v4u g0;
    g0.x = 1u;                                        // count=1, no gather
    g0.y = ldsOff;                                    // lds_addr
    g0.z = (unsigned)(ga & 0xffffffffu);              // global_addr[31:0]
    g0.w = (unsigned)((ga >> 32) & 0x01ffffffu) | (2u << 30);  // addr[56:32] | type=2
    // d0: data_size=4B(2), pad_enable, pad_interval=256 dwords(7), pad_amount=2 dwords(1)
    v8i g1 = { (int)((2u << 16) | (1u << 20) | (7u << 22) | (1u << 25)),
               (int)(512u << 16),     // tensor_dim0[15:0]=512
               (int)(16u << 16),      // tensor_dim0 hi=0 | tensor_dim1[15:0]=16
               (int)(512u << 16),     // tensor_dim1 hi=0 | tile_dim0=512
               16,                    // tile_dim1=16, tile_dim2=0
               512,                   // tensor_dim0_stride[31:0]=512
               0, 0 };
    v4i gz = {0, 0, 0, 0};
    if (wave == 0) {
#if defined(__clang_major__) && __clang_major__ >= 23
      v8i gz8 = {0, 0, 0, 0, 0, 0, 0, 0};
      __builtin_amdgcn_tensor_load_to_lds(g0, g1, gz, gz, gz8, 0);
#else
      __builtin_amdgcn_tensor_load_to_lds(g0, g1, gz, gz, 0);
#endif
      __builtin_amdgcn_s_wait_tensorcnt(0);
    }
  }
  __syncthreads();

  const float* bp0 = x + (size_t)(colBase +  0 + l) * DD + 2 * hv;
  const float* bp1 = x + (size_t)(colBase + 16 + l) * DD + 2 * hv;
  const float* bp2 = x + (size_t)(colBase + 32 + l) * DD + 2 * hv;
  const float* bp3 = x + (size_t)(colBase + 48 + l) * DD + 2 * hv;

  v8f acc0 = {}, acc1 = {}, acc2 = {}, acc3 = {};
  for (int k = 0; k < DD; k += 4) {
    int c = k + 2 * hv;                               // column of A fragment
    int ci = c + ((c >> 8) << 1);                     // account for TDM pad (always even)
    v2f a  = *(const v2f*)(sA + l * AStride + ci);    // ds_load_b64, conflict-free
    v2f b0 = *(const v2f*)(bp0 + k);
    v2f b1 = *(const v2f*)(bp1 + k);
    v2f b2 = *(const v2f*)(bp2 + k);
    v2f b3 = *(const v2f*)(bp3 + k);
    acc0 = __builtin_amdgcn_wmma_f32_16x16x4_f32(false, a, false, b0, (short)0, acc0, false, false);
    acc1 = __builtin_amdgcn_wmma_f32_16x16x4_f32(false, a, false, b1, (short)0, acc1, false, false);
    acc2 = __builtin_amdgcn_wmma_f32_16x16x4_f32(false, a, false, b2, (short)0, acc2, false, false);
    acc3 = __builtin_amdgcn_wmma_f32_16x16x4_f32(false, a, false, b3, (short)0, acc3, false, false);
  }

  float si[8];
#pragma unroll
  for (int e = 0; e < 8; ++e) si[e] = sq[rowBase + 8 * hv + e];

#define EPILOGUE(ACC, C)                                          \
  {                                                               \
    int j = colBase + (C) * 16 + l;                               \
    float sj = sq[j];                                             \
    _Pragma("unroll")                                             \
    for (int e = 0; e < 8; ++e) {                                 \
      int i = rowBase + 8 * hv + e;                               \
      float d2 = si[e] + sj - 2.0f * ACC[e];                      \
      d2 = fmaxf(d2, 1e-12f);                                     \
      if (i == j) d2 = BIGF;                                      \
      dist[(size_t)i * NN + j] = sqrtf(d2);                       \
    }                                                             \
  }
  EPILOGUE(acc0, 0) EPILOGUE(acc1, 1) EPILOGUE(acc2, 2) EPILOGUE(acc3, 3)
#undef EPILOGUE
}

// ---------------- Prim MST: fused relax+argmin, one WGP per matrix ----------------
// blockIdx.x selects the matrix; launch with grid=2 to run both serial chains
// concurrently on separate WGPs.
__global__ __launch_bounds__(1024) void prim_kernel(const float* __restrict__ distA,
                                                    const float* __restrict__ distB,
                                                    float* __restrict__ deaths) {
  const float* dist = (blockIdx.x == 0) ? distA : distB;
  float* dth = deaths + (size_t)blockIdx.x * NN;

  __shared__ float md[NN];
  __shared__ unsigned char flags[NN];
  __shared__ unsigned long long red[32];
  __shared__ int selj;

  const int tid  = threadIdx.x;
  const int lane = tid & 31;
  const int wid  = tid >> 5;

  // init from dist row 0; vertex 0 in tree; also produce first local argmin
  unsigned long long best = ~0ull;
#pragma unroll
  for (int t = 0; t < NN / 1024; ++t) {
    int k = tid + t * 1024;
    float v = (k == 0) ? BIGF : dist[k];
    md[k] = v;
    flags[k] = (k == 0) ? 1 : 0;
    unsigned long long p = ((unsigned long long)__float_as_uint(v) << 32) | (unsigned)k;
    best = (p < best) ? p : best;
  }
  __syncthreads();

  for (int step = 0; step < NN - 1; ++step) {
    // block argmin of previous pass's local minima
#pragma unroll
    for (int off = 16; off > 0; off >>= 1) {
      unsigned long long o = shfl_xor_u64(best, off);
      best = (o < best) ? o : best;
    }
    if (lane == 0) red[wid] = best;
    __syncthreads();
    if (wid == 0) {
      unsigned long long b = red[lane];
#pragma unroll
      for (int off = 16; off > 0; off >>= 1) {
        unsigned long long o = shfl_xor_u64(b, off);
        b = (o < b) ? o : b;
      }
      if (lane == 0) {
        int j = (int)(b & 0xffffffffu);
        dth[step] = __uint_as_float((unsigned)(b >> 32));   // H0 death
        flags[j] = 1;
        md[j] = BIGF;
        selj = j;
      }
    }
    __syncthreads();
    // fused: relax against dist[selj] AND compute next local argmin in one sweep
    const float* row = dist + (size_t)selj * NN;
    best = ~0ull;
#pragma unroll
    for (int t = 0; t < NN / 1024; ++t) {
      int k = tid + t * 1024;
      if (!flags[k]) {
        float v = fminf(md[k], row[k]);
        md[k] = v;
        unsigned long long p = ((unsigned long long)__float_as_uint(v) << 32) | (unsigned)k;
        best = (p < best) ? p : best;
      }
    }
    // no trailing barrier needed: md[k] is thread-private under this partition;
    // cross-thread pokes (flags[j], md[j], selj, red) are ordered by the two
    // barriers above.
  }
  if (tid == 0) dth[NN - 1] = __builtin_inff();   // pad slot for bitonic sort
}

// ---------------- bitonic sort, one diagram per block ----------------
__global__ __launch_bounds__(1024) void sort_kernel(float* __restrict__ deaths) {
  __shared__ float s[NN];
  float* a = deaths + (size_t)blockIdx.x * NN;
#pragma unroll
  for (int t = 0; t < NN / 1024; ++t) s[threadIdx.x + t * 1024] = a[threadIdx.x + t * 1024];
  __syncthreads();
  for (int k = 2; k <= NN; k <<= 1) {
    for (int j = k >> 1; j > 0; j >>= 1) {
#pragma unroll
      for (int t = 0; t < NN / 1024; ++t) {
        int i = threadIdx.x + t * 1024;
        int ixj = i ^ j;
        if (ixj > i) {
          bool up = ((i & k) == 0);
          float vi = s[i], vj = s[ixj];
          if ((vi > vj) == up) { s[i] = vj; s[ixj] = vi; }
        }
      }
      __syncthreads();
    }
  }
#pragma unroll
  for (int t = 0; t < NN / 1024; ++t) a[threadIdx.x + t * 1024] = s[threadIdx.x + t * 1024];
}

// ---------------- L2 partial sums (deterministic tree) ----------------
__global__ __launch_bounds__(256) void l2_partial_kernel(const float* __restrict__ x1,
                                                         const float* __restrict__ x2,
                                                         float* __restrict__ partials) {
  __shared__ float r[256];
  int gid = blockIdx.x * 256 + threadIdx.x;
  float s = 0.f;
  for (int i = gid; i < NN * DD; i += 256 * 256) {
    float d = x1[i] - x2[i];
    s = fmaf(d, d, s);
  }
  r[threadIdx.x] = s;
  __syncthreads();
  for (int off = 128; off > 0; off >>= 1) {
    if (threadIdx.x < off) r[threadIdx.x] += r[threadIdx.x + off];
    __syncthreads();
  }
  if (threadIdx.x == 0) partials[blockIdx.x] = r[0];
}

// ---------------- final: L2 total + Wasserstein-1 ----------------
__global__ __launch_bounds__(256) void final_kernel(const float* __restrict__ partials,
                                                    const float* __restrict__ sd1,
                                                    const float* __restrict__ sd2,
                                                    float* __restrict__ out) {
  __shared__ float r[256];
  const int tid = threadIdx.x;
  r[tid] = partials[tid];
  __syncthreads();
  for (int off = 128; off > 0; off >>= 1) {
    if (tid < off) r[tid] += r[tid + off];
    __syncthreads();
  }
  float l2tot = r[0];
  __syncthreads();
  float wc = 0.f;
  for (int i = tid; i < NN - 1; i += 256) {
    float a = sd1[i], b = sd2[i];
    wc += fminf(fabsf(a - b), 0.5f * (a + b));
  }
  r[tid] = wc;
  __syncthreads();
  for (int off = 128; off > 0; off >>= 1) {
    if (tid < off) r[tid] += r[tid + off];
    __syncthreads();
  }
  if (tid == 0) out[0] = l2tot / (float)(NN * DD) + r[0];
}

extern "C" void kernel_launch(void* const* d_in, const int* in_sizes, int n_in,
                              void* d_out, int out_size, void* d_ws, size_t ws_size,
                              hipStream_t stream) {
  const float* x1 = (const float*)d_in[0];
  const float* x2 = (const float*)d_in[1];
  float* out = (float*)d_out;
  float* ws = (float*)d_ws;

  const size_t needDual = ((size_t)2 * NN * NN + 2 * NN + 2 * NN + 256) * sizeof(float);

  if (ws_size >= needDual) {
    // dual buffers: both Prim chains run concurrently (grid=2, one WGP each)
    float* dA = ws;
    float* dB = dA + (size_t)NN * NN;
    float* sq = dB + (size_t)NN * NN;          // sq1 | sq2
    float* deaths = sq + 2 * NN;               // two padded 4096 arrays
    float* partials = deaths + 2 * NN;

    sqnorm_kernel<<<NN / 8, 256, 0, stream>>>(x1, sq);
    sqnorm_kernel<<<NN / 8, 256, 0, stream>>>(x2, sq + NN);
    dist_kernel<<<dim3(8, NN / 16), 256, 0, stream>>>(x1, sq, dA);
    dist_kernel<<<dim3(8, NN / 16), 256, 0, stream>>>(x2, sq + NN, dB);
    prim_kernel<<<2, 1024, 0, stream>>>(dA, dB, deaths);
    l2_partial_kernel<<<256, 256, 0, stream>>>(x1, x2, partials);
    sort_kernel<<<2, 1024, 0, stream>>>(deaths);
    final_kernel<<<1, 256, 0, stream>>>(partials, deaths, deaths + NN, out);
  } else {
    // sequential fallback: one 64 MB distance buffer reused
    float* dA = ws;
    float* sq = dA + (size_t)NN * NN;
    float* deaths = sq + NN;
    float* partials = deaths + 2 * NN;

    for (int m = 0; m < 2; ++m) {
      const float* x = (m == 0) ? x1 : x2;
      sqnorm_kernel<<<NN / 8, 256, 0, stream>>>(x, sq);
      dist_kernel<<<dim3(8, NN / 16), 256, 0, stream>>>(x, sq, dA);
      prim_kernel<<<1, 1024, 0, stream>>>(dA, dA, deaths + (size_t)m * NN);
    }
    l2_partial_kernel<<<256, 256, 0, stream>>>(x1, x2, partials);
    sort_kernel<<<2, 1024, 0, stream>>>(deaths);
    final_kernel<<<1, 256, 0, stream>>>(partials, deaths, deaths + NN, out);
  }
}